// BahdanauAttention_50663434223729
// MI455X (gfx1250) — compile-verified
//
#include <hip/hip_runtime.h>
#include <math.h>

typedef __attribute__((ext_vector_type(2))) float v2f;
typedef __attribute__((ext_vector_type(8))) float v8f;

#define HDIM 2048
#define SDIM 4096
#define LSTRIDE (HDIM + 4)   // LDS row pad: half-waves map to all 64 banks

static __device__ __forceinline__ v8f wmma_f32(v2f a, v2f b, v8f c) {
  return __builtin_amdgcn_wmma_f32_16x16x4_f32(false, a, false, b, (short)0, c,
                                               false, false);
}

// branch-free tanh: 1 - 2/(e^{2x}+1); e->inf => 1, e->0 => -1 (no NaN)
static __device__ __forceinline__ float tanh_fast(float x) {
  const float e = __expf(2.0f * x);
  return 1.0f - 2.0f / (e + 1.0f);
}

// ---------------------------------------------------------------------------
// Kernel 1: whb[h] = (W_h @ decoder_hidden)[h] + b[h]      (GEMV, mem-bound)
// ---------------------------------------------------------------------------
__global__ __launch_bounds__(256) void whb_kernel(const float* __restrict__ W_h,
                                                  const float* __restrict__ dh,
                                                  const float* __restrict__ b,
                                                  float* __restrict__ whb) {
  const int wave = threadIdx.x >> 5;
  const int lane = threadIdx.x & 31;
  const int row  = blockIdx.x * 8 + wave;
  const float* wrow = W_h + (size_t)row * HDIM;
  float sum = 0.f;
  #pragma unroll 4
  for (int k = lane; k < HDIM; k += 32) sum += wrow[k] * dh[k];
  #pragma unroll
  for (int m = 16; m >= 1; m >>= 1) sum += __shfl_xor(sum, m, 32);
  if (lane == 0) whb[row] = sum + b[row];
}

// ---------------------------------------------------------------------------
// Kernel 2: fused scores[s] = sum_h tanh((enc @ W_e^T)[s,h] + whb[h]) * v[h]
//
// K-chunk loop outermost; each wave keeps all 16 h-tiles' accumulators live
// (8 pairs x 2 x v8f = 128 VGPRs). Per K-chunk (16 floats of K):
//   - A chunk: ds_load_2addr_b64 pairs from the LDS-staged s-tile, reused by
//     all 64 WMMAs of the chunk,
//   - B (W_e): two-deep register pipeline over the 8 pairs (parity = p&1);
//     pair p's 8 loads issue before pair p-1's 8 WMMAs -> waits tolerate a
//     full chunk of outstanding loads behind the fp32 matrix pipe.
// Epilogue: per-lane accumulation across pairs (reduction is linear), one
// 16-lane shuffle reduction at the very end -> 32 shuffles total.
// ---------------------------------------------------------------------------
__global__ __launch_bounds__(256) void energy_scores_kernel(
    const float* __restrict__ enc, const float* __restrict__ W_e,
    const float* __restrict__ whb, const float* __restrict__ v,
    float* __restrict__ scores) {
  extern __shared__ float s_enc[];       // 16 * LSTRIDE floats (~128 KB)
  __shared__ float s_scores[16];

  const int tid  = threadIdx.x;
  const int wave = tid >> 5;
  const int lane = tid & 31;
  const int lm   = lane & 15;
  const int hi   = lane >> 4;            // 0: lanes 0-15, 1: lanes 16-31
  const int s0   = blockIdx.x * 16;

  if (tid < 16) s_scores[tid] = 0.f;

  // cooperative stage: 16 x HDIM floats, float4 on both sides
  for (int i = tid; i < 16 * (HDIM / 4); i += 256) {
    const int row  = i >> 9;             // HDIM/4 = 512 float4 per row
    const int col4 = i & 511;
    const float4 d =
        *(const float4*)(enc + (size_t)(s0 + row) * HDIM + (col4 << 2));
    *(float4*)(&s_enc[row * LSTRIDE + (col4 << 2)]) = d;
  }
  __syncthreads();

  const float* aRow = s_enc + lm * LSTRIDE + 2 * hi;
  // wave owns h-tiles [wave*16, wave*16+16); pair p = tiles 2p, 2p+1
  const float* bBase = W_e + (size_t)(wave * 256 + lm) * HDIM + 2 * hi;

  v8f acc[8][2];
  #pragma unroll
  for (int p = 0; p < 8; ++p) {
    acc[p][0] = (v8f){};
    acc[p][1] = (v8f){};
  }

  v2f Bb[2][8];   // [parity][tile*4 + j]
  #pragma unroll
  for (int j = 0; j < 4; ++j) {          // preload position (c=0, p=0)
    Bb[0][j]     = *(const v2f*)(bBase + 4 * j);
    Bb[0][4 + j] = *(const v2f*)(bBase + (size_t)16 * HDIM + 4 * j);
  }

  #pragma unroll 1
  for (int c = 0; c < HDIM / 16; ++c) {
    const int k = c << 4;

    v2f a[4];                            // A chunk: 8 VGPRs from LDS
    #pragma unroll
    for (int j = 0; j < 4; ++j) a[j] = *(const v2f*)(aRow + k + 4 * j);

    #pragma unroll
    for (int p = 0; p < 8; ++p) {
      const int cur = p & 1, nxt = cur ^ 1;

      // prefetch next position's B chunk (pair p+1, or (c+1, p=0))
      if (p < 7) {
        const float* nb = bBase + (size_t)(p + 1) * 32 * HDIM + k;
        #pragma unroll
        for (int j = 0; j < 4; ++j) {
          Bb[nxt][j]     = *(const v2f*)(nb + 4 * j);
          Bb[nxt][4 + j] = *(const v2f*)(nb + (size_t)16 * HDIM + 4 * j);
        }
      } else {
        const int k2 = (c < HDIM / 16 - 1) ? (k + 16) : 0;  // wrap: no OOB
        const float* nb = bBase + k2;
        #pragma unroll
        for (int j = 0; j < 4; ++j) {
          Bb[nxt][j]     = *(const v2f*)(nb + 4 * j);
          Bb[nxt][4 + j] = *(const v2f*)(nb + (size_t)16 * HDIM + 4 * j);
        }
      }

      // consume current position's B chunk
      #pragma unroll
      for (int j = 0; j < 4; ++j) {
        acc[p][0] = wmma_f32(a[j], Bb[cur][j], acc[p][0]);
        acc[p][1] = wmma_f32(a[j], Bb[cur][4 + j], acc[p][1]);
      }
    }
  }

  // epilogue: per-lane accumulation over pairs (no shuffles in this loop)
  float lacc[8];
  #pragma unroll
  for (int r = 0; r < 8; ++r) lacc[r] = 0.f;

  #pragma unroll
  for (int p = 0; p < 8; ++p) {
    const int h0 = wave * 256 + p * 32;
    const float wb0 = whb[h0 + lm],      vv0 = v[h0 + lm];
    const float wb1 = whb[h0 + 16 + lm], vv1 = v[h0 + 16 + lm];
    #pragma unroll
    for (int r = 0; r < 8; ++r) {
      lacc[r] += tanh_fast(acc[p][0][r] + wb0) * vv0 +
                 tanh_fast(acc[p][1][r] + wb1) * vv1;
    }
  }

  // single 16-lane reduction per r: lane 0 -> m=r, lane 16 -> m=r+8
  #pragma unroll
  for (int r = 0; r < 8; ++r) {
    float t = lacc[r];
    t += __shfl_xor(t, 8, 32);
    t += __shfl_xor(t, 4, 32);
    t += __shfl_xor(t, 2, 32);
    t += __shfl_xor(t, 1, 32);
    if (lm == 0) atomicAdd(&s_scores[r + 8 * hi], t);
  }
  __syncthreads();
  if (tid < 16) scores[s0 + tid] = s_scores[tid];
}

// ---------------------------------------------------------------------------
// Kernel 3: softmax over scores (S=4096), one block of 1024 threads.
// Also zeroes the context region of d_out (accumulated atomically later).
// ---------------------------------------------------------------------------
__global__ __launch_bounds__(1024) void softmax_kernel(
    const float* __restrict__ scores, float* __restrict__ out) {
  __shared__ float red[32];
  __shared__ float stat[2];
  const int tid = threadIdx.x;

  out[tid] = 0.f;
  out[tid + 1024] = 0.f;

  float vals[4];
  float m = -INFINITY;
  #pragma unroll
  for (int i = 0; i < 4; ++i) {
    vals[i] = scores[tid + i * 1024];
    m = fmaxf(m, vals[i]);
  }
  #pragma unroll
  for (int o = 16; o >= 1; o >>= 1) m = fmaxf(m, __shfl_xor(m, o, 32));
  if ((tid & 31) == 0) red[tid >> 5] = m;
  __syncthreads();
  if (tid < 32) {
    float x = red[tid];
    #pragma unroll
    for (int o = 16; o >= 1; o >>= 1) x = fmaxf(x, __shfl_xor(x, o, 32));
    if (tid == 0) stat[0] = x;
  }
  __syncthreads();
  m = stat[0];

  float lsum = 0.f;
  #pragma unroll
  for (int i = 0; i < 4; ++i) {
    vals[i] = __expf(vals[i] - m);
    lsum += vals[i];
  }
  #pragma unroll
  for (int o = 16; o >= 1; o >>= 1) lsum += __shfl_xor(lsum, o, 32);
  if ((tid & 31) == 0) red[tid >> 5] = lsum;
  __syncthreads();
  if (tid < 32) {
    float x = red[tid];
    #pragma unroll
    for (int o = 16; o >= 1; o >>= 1) x += __shfl_xor(x, o, 32);
    if (tid == 0) stat[1] = x;
  }
  __syncthreads();
  const float inv = 1.0f / stat[1];
  #pragma unroll
  for (int i = 0; i < 4; ++i) out[HDIM + tid + i * 1024] = vals[i] * inv;
}

// ---------------------------------------------------------------------------
// Kernel 4: context[h] = sum_s attn[s] * enc[s,h]   (mem-bound, 32 MB read)
// ---------------------------------------------------------------------------
__global__ __launch_bounds__(256) void context_kernel(
    const float* __restrict__ enc, const float* __restrict__ attn,
    float* __restrict__ context) {
  __shared__ float s_attn[256];
  const int t = threadIdx.x;
  const int sbase = blockIdx.x * 256;
  s_attn[t] = attn[sbase + t];
  __syncthreads();

  for (int hb = 0; hb < HDIM; hb += 256) {
    const int h = hb + t;
    float sum = 0.f;
    #pragma unroll 4
    for (int s = 0; s < 256; ++s)
      sum += s_attn[s] * enc[(size_t)(sbase + s) * HDIM + h];
    atomicAdd(&context[h], sum);
  }
}

// ---------------------------------------------------------------------------
extern "C" void kernel_launch(void* const* d_in, const int* in_sizes, int n_in,
                              void* d_out, int out_size, void* d_ws, size_t ws_size,
                              hipStream_t stream) {
  const float* dh  = (const float*)d_in[0];  // (H,1)
  const float* enc = (const float*)d_in[1];  // (S,H,1) -> (S,H)
  const float* W_h = (const float*)d_in[2];  // (H,H)
  const float* W_e = (const float*)d_in[3];  // (H,H)
  const float* b   = (const float*)d_in[4];  // (H,1)
  const float* v   = (const float*)d_in[5];  // (1,H)

  float* out    = (float*)d_out;             // [0,H): context, [H,H+S): attn
  float* whb    = (float*)d_ws;              // H floats
  float* scores = whb + HDIM;                // S floats

  const size_t lds_bytes = (size_t)16 * LSTRIDE * sizeof(float);  // ~128 KB

  whb_kernel<<<HDIM / 8, 256, 0, stream>>>(W_h, dh, b, whb);
  energy_scores_kernel<<<SDIM / 16, 256, lds_bytes, stream>>>(enc, W_e, whb, v,
                                                              scores);
  softmax_kernel<<<1, 1024, 0, stream>>>(scores, out);
  context_kernel<<<SDIM / 256, 256, 0, stream>>>(enc, out + HDIM, out);
}